// GNN_45801531244885
// MI455X (gfx1250) — compile-verified
//
#include <hip/hip_runtime.h>
#include <hip/hip_bf16.h>

typedef float v2f __attribute__((ext_vector_type(2)));
typedef float v8f __attribute__((ext_vector_type(8)));

__device__ __forceinline__ void atomAddF(float* p, float v) {
  __hip_atomic_fetch_add(p, v, __ATOMIC_RELAXED, __HIP_MEMORY_SCOPE_AGENT);
}
__device__ __forceinline__ float leaky02(float x) { return x > 0.f ? x : 0.2f * x; }
// order-preserving float<->uint encoding for atomicMax on floats
__device__ __forceinline__ unsigned encf(float f) {
  unsigned u = __float_as_uint(f);
  return (u & 0x80000000u) ? ~u : (u | 0x80000000u);
}
__device__ __forceinline__ float decf(unsigned e) {
  return __uint_as_float((e & 0x80000000u) ? (e & 0x7fffffffu) : ~e);
}

// ---------------- WMMA fp32 GEMM: C[M,Nn] = A[M,K] @ B[K,Nn] ----------------
// one wave32 per 16x16 output tile, K stepped by 4 via V_WMMA_F32_16X16X4_F32
__global__ void k_gemm_wmma(const float* __restrict__ A, const float* __restrict__ B,
                            float* __restrict__ C, int M, int Nn, int K) {
  int wave = (blockIdx.x * blockDim.x + threadIdx.x) >> 5;
  int lane = threadIdx.x & 31;
  int ntiles = Nn >> 4;
  int tiles = (M >> 4) * ntiles;
  if (wave >= tiles) return;                 // wave-uniform exit, EXEC stays all-1
  int tm = (wave / ntiles) << 4;
  int tn = (wave % ntiles) << 4;
  int row = lane & 15;                       // A row / B,C col within tile
  int half = lane >> 4;                      // K/M half selector
  v8f acc = {0.f, 0.f, 0.f, 0.f, 0.f, 0.f, 0.f, 0.f};
  const float* Arow = A + (size_t)(tm + row) * K + 2 * half;
  const float* Bcol = B + (size_t)tn + row;
  for (int kb = 0; kb < K; kb += 4) {
    v2f a, b;
    a.x = Arow[kb];
    a.y = Arow[kb + 1];
    b.x = Bcol[(size_t)(kb + 2 * half) * Nn];
    b.y = Bcol[(size_t)(kb + 2 * half + 1) * Nn];
    acc = __builtin_amdgcn_wmma_f32_16x16x4_f32(false, a, false, b, (short)0, acc,
                                                false, false);
  }
  for (int v = 0; v < 8; ++v)
    C[(size_t)(tm + v + 8 * half) * Nn + tn + row] = acc[v];
}

// ---------------- degree / norm ----------------
__global__ void k_fill(float* p, float v, int total) {
  int i = blockIdx.x * blockDim.x + threadIdx.x;
  if (i < total) p[i] = v;
}
__global__ void k_deg_edges(float* deg, const int* __restrict__ dst, int e) {
  int i = blockIdx.x * blockDim.x + threadIdx.x;
  if (i < e) atomAddF(deg + dst[i], 1.f);
}
__global__ void k_dinv(float* d, int n) {
  int i = blockIdx.x * blockDim.x + threadIdx.x;
  if (i < n) d[i] = __frsqrt_rn(d[i]);       // deg >= 1 always (self loops)
}

// ---------------- GCN aggregation ----------------
__global__ void k_gcn_init(float* __restrict__ out, const float* __restrict__ xw,
                           const float* __restrict__ dis, int D, int total) {
  int i = blockIdx.x * blockDim.x + threadIdx.x;
  if (i >= total) return;
  int n = i / D;
  float s = dis[n];
  out[i] = s * s * xw[i];                    // self-loop contribution
}
__global__ void k_gcn_edges(float* __restrict__ out, const float* __restrict__ xw,
                            const float* __restrict__ dis, const int* __restrict__ src,
                            const int* __restrict__ dst, int D, int e) {
  int wave = (blockIdx.x * blockDim.x + threadIdx.x) >> 5;
  int lane = threadIdx.x & 31;
  if (wave >= e) return;
  int s = src[wave], d = dst[wave];
  float norm = dis[s] * dis[d];
  int per = D >> 5;                          // 2 (D=64) or 4 (D=128) dims/lane
  const float* xs = xw + (size_t)s * D + lane * per;
  float* od = out + (size_t)d * D + lane * per;
  __builtin_prefetch(xs, 0, 0);
  for (int j = 0; j < per; ++j) atomAddF(od + j, norm * xs[j]);
}

// ---------------- bias + optional relu ----------------
__global__ void k_bias_act(float* __restrict__ buf, const float* __restrict__ bias,
                           int dmask, int relu, int total) {
  int i = blockIdx.x * blockDim.x + threadIdx.x;
  if (i >= total) return;
  float v = buf[i] + bias[i & dmask];
  buf[i] = (relu && v < 0.f) ? 0.f : v;
}

// ---------------- GAT pieces (HEADS=4, CH=64, F=256) ----------------
__global__ void k_att_scores(const float* __restrict__ xl, const float* __restrict__ asrc,
                             const float* __restrict__ adst, float* __restrict__ als,
                             float* __restrict__ ald, int n) {
  int i = blockIdx.x * blockDim.x + threadIdx.x;
  if (i >= n * 4) return;
  int nn = i >> 2, h = i & 3;
  const float* xr = xl + (size_t)nn * 256 + h * 64;
  const float* as_ = asrc + h * 64;
  const float* ad_ = adst + h * 64;
  float s1 = 0.f, s2 = 0.f;
  for (int c = 0; c < 64; ++c) {
    float v = xr[c];
    s1 += v * as_[c];
    s2 += v * ad_[c];
  }
  als[i] = s1;
  ald[i] = s2;
}
__global__ void k_amax_init(const float* __restrict__ als, const float* __restrict__ ald,
                            unsigned* __restrict__ amax, int total) {
  int i = blockIdx.x * blockDim.x + threadIdx.x;
  if (i < total) amax[i] = encf(leaky02(als[i] + ald[i]));   // self edge
}
__global__ void k_amax_edges(const float* __restrict__ als, const float* __restrict__ ald,
                             unsigned* __restrict__ amax, const int* __restrict__ src,
                             const int* __restrict__ dst, int e4) {
  int t = blockIdx.x * blockDim.x + threadIdx.x;
  if (t >= e4) return;
  int ee = t >> 2, h = t & 3;
  int s = src[ee], d = dst[ee];
  float alpha = leaky02(als[s * 4 + h] + ald[d * 4 + h]);
  atomicMax(amax + d * 4 + h, encf(alpha));
}
__global__ void k_den_init(const float* __restrict__ als, const float* __restrict__ ald,
                           const unsigned* __restrict__ amax, float* __restrict__ den,
                           int total) {
  int i = blockIdx.x * blockDim.x + threadIdx.x;
  if (i < total) den[i] = __expf(leaky02(als[i] + ald[i]) - decf(amax[i]));
}
__global__ void k_den_edges(const float* __restrict__ als, const float* __restrict__ ald,
                            const unsigned* __restrict__ amax, float* __restrict__ den,
                            const int* __restrict__ src, const int* __restrict__ dst, int e4) {
  int t = blockIdx.x * blockDim.x + threadIdx.x;
  if (t >= e4) return;
  int ee = t >> 2, h = t & 3;
  int s = src[ee], d = dst[ee];
  float alpha = leaky02(als[s * 4 + h] + ald[d * 4 + h]);
  atomAddF(den + d * 4 + h, __expf(alpha - decf(amax[d * 4 + h])));
}
__global__ void k_gat_init(float* __restrict__ out, const float* __restrict__ xl,
                           const float* __restrict__ als, const float* __restrict__ ald,
                           const unsigned* __restrict__ amax, const float* __restrict__ den,
                           int total) {
  int i = blockIdx.x * blockDim.x + threadIdx.x;
  if (i >= total) return;
  int nn = i >> 8, h = (i & 255) >> 6;
  int nh = nn * 4 + h;
  float alpha = leaky02(als[nh] + ald[nh]);
  float coef = __expf(alpha - decf(amax[nh])) / den[nh];
  out[i] = coef * xl[i];                      // self-loop term
}
__global__ void k_gat_edges(float* __restrict__ out, const float* __restrict__ xl,
                            const float* __restrict__ als, const float* __restrict__ ald,
                            const unsigned* __restrict__ amax, const float* __restrict__ den,
                            const int* __restrict__ src, const int* __restrict__ dst, int e) {
  int wave = (blockIdx.x * blockDim.x + threadIdx.x) >> 5;
  int lane = threadIdx.x & 31;
  if (wave >= e) return;
  int s = src[wave], d = dst[wave];
  int h = lane >> 3;                          // 8 lanes per head, 8 dims per lane
  int sh = s * 4 + h, dh = d * 4 + h;
  float alpha = leaky02(als[sh] + ald[dh]);
  float coef = __expf(alpha - decf(amax[dh])) / den[dh];
  const float* xs = xl + (size_t)s * 256 + lane * 8;
  float* od = out + (size_t)d * 256 + lane * 8;
  __builtin_prefetch(xs, 0, 0);
  for (int j = 0; j < 8; ++j) atomAddF(od + j, coef * xs[j]);
}

// ---------------- host launch ----------------
extern "C" void kernel_launch(void* const* d_in, const int* in_sizes, int n_in,
                              void* d_out, int out_size, void* d_ws, size_t ws_size,
                              hipStream_t stream) {
  const float* x   = (const float*)d_in[0];
  const int*   ei  = (const int*)d_in[1];
  const float* W1  = (const float*)d_in[3];
  const float* b1  = (const float*)d_in[4];
  const float* Wa1 = (const float*)d_in[5];
  const float* as1 = (const float*)d_in[6];
  const float* ad1 = (const float*)d_in[7];
  const float* ba1 = (const float*)d_in[8];
  const float* Wa2 = (const float*)d_in[9];
  const float* as2 = (const float*)d_in[10];
  const float* ad2 = (const float*)d_in[11];
  const float* ba2 = (const float*)d_in[12];
  const float* W2  = (const float*)d_in[13];
  const float* b2  = (const float*)d_in[14];
  const int n = in_sizes[0] / 128;
  const int e = in_sizes[1] / 2;
  const int* src = ei;
  const int* dst = ei + e;

  char* w = (char*)d_ws;
  auto carve = [&](size_t bytes) -> void* {
    void* p = (void*)w;
    w += (bytes + 255) & ~(size_t)255;
    return p;
  };
  float*    dis  = (float*)carve((size_t)n * 4);
  float*    xw   = (float*)carve((size_t)n * 256 * 4);
  float*    hb   = (float*)carve((size_t)n * 256 * 4);
  float*    als  = (float*)carve((size_t)n * 4 * 4);
  float*    ald  = (float*)carve((size_t)n * 4 * 4);
  unsigned* amax = (unsigned*)carve((size_t)n * 4 * 4);
  float*    den  = (float*)carve((size_t)n * 4 * 4);
  (void)ws_size; (void)n_in; (void)out_size;

  const int B = 256;
  auto nb = [](int total, int blk) { return (total + blk - 1) / blk; };
  auto gemm = [&](const float* A, const float* Bm, float* C, int M, int Nn, int K) {
    int tiles = (M / 16) * (Nn / 16);
    k_gemm_wmma<<<nb(tiles * 32, B), B, 0, stream>>>(A, Bm, C, M, Nn, K);
  };
  int ewaveBlocks = nb(e * 32, B);

  // degree -> D^-1/2  (self loops included)
  k_fill<<<nb(n, B), B, 0, stream>>>(dis, 1.f, n);
  k_deg_edges<<<nb(e, B), B, 0, stream>>>(dis, dst, e);
  k_dinv<<<nb(n, B), B, 0, stream>>>(dis, n);

  // ---- Layer 1: GCN(128->64) + ReLU ----
  gemm(x, W1, xw, n, 64, 128);
  k_gcn_init<<<nb(n * 64, B), B, 0, stream>>>(hb, xw, dis, 64, n * 64);
  k_gcn_edges<<<ewaveBlocks, B, 0, stream>>>(hb, xw, dis, src, dst, 64, e);
  k_bias_act<<<nb(n * 64, B), B, 0, stream>>>(hb, b1, 63, 1, n * 64);

  // ---- GAT layers ----
  auto gat = [&](int K, const float* Wa, const float* asc, const float* adc,
                 const float* ba) {
    gemm(hb, Wa, xw, n, 256, K);
    k_att_scores<<<nb(n * 4, B), B, 0, stream>>>(xw, asc, adc, als, ald, n);
    k_amax_init<<<nb(n * 4, B), B, 0, stream>>>(als, ald, amax, n * 4);
    k_amax_edges<<<nb(e * 4, B), B, 0, stream>>>(als, ald, amax, src, dst, e * 4);
    k_den_init<<<nb(n * 4, B), B, 0, stream>>>(als, ald, amax, den, n * 4);
    k_den_edges<<<nb(e * 4, B), B, 0, stream>>>(als, ald, amax, den, src, dst, e * 4);
    k_gat_init<<<nb(n * 256, B), B, 0, stream>>>(hb, xw, als, ald, amax, den, n * 256);
    k_gat_edges<<<ewaveBlocks, B, 0, stream>>>(hb, xw, als, ald, amax, den, src, dst, e);
    k_bias_act<<<nb(n * 256, B), B, 0, stream>>>(hb, ba, 255, 1, n * 256);
  };
  gat(64,  Wa1, as1, ad1, ba1);   // Layer 2: GAT(64 -> 4x64) + ReLU
  gat(256, Wa2, as2, ad2, ba2);   // Layer 3: GAT(256 -> 4x64) + ReLU

  // ---- Layer 4: GCN(256->128), no ReLU ----
  float* out = (float*)d_out;
  gemm(hb, W2, xw, n, 128, 256);
  k_gcn_init<<<nb(n * 128, B), B, 0, stream>>>(out, xw, dis, 128, n * 128);
  k_gcn_edges<<<ewaveBlocks, B, 0, stream>>>(out, xw, dis, src, dst, 128, e);
  k_bias_act<<<nb(n * 128, B), B, 0, stream>>>(out, b2, 127, 0, n * 128);
}